// Messages_21251498181146
// MI455X (gfx1250) — compile-verified
//
#include <hip/hip_runtime.h>

// ---------------------------------------------------------------------------
// Equivariant message passing (9 low-rank tensor products + scatter-add)
// for MI455X / gfx1250, built on V_WMMA_F32_16X16X4_F32 (full f32 precision,
// matching the f32 reference).
//
// Factorization used (per TP, per edge):
//   g[p]      = wr @ rad                                  (32x8 GEMM)
//   L[p,a]    = wl @ x_dst[:,a]                           (32xdim_l GEMM)  <- dominant
//   m[p,a']   = sum_kidx L[p,(a',kidx)] * rh^{ox k}[kidx]
//   out[o,a'] = sum_p wo[o,p] * g[p]*m[p,a']              (dim_out x 32 GEMM) <- dominant
//   psi[o,(a',b')] = out[o,a'] * rh^{ox (p_r-k)}[b']  -> atomicAdd @ src node
//
// One wave per 16-edge tile (N dim of every WMMA = 16 edges).
// C/D layout per ISA: vgpr v holds (M=v, lanes0-15) and (M=v+8, lanes16-31),
// N = lane%16. A 16x4 layout per ISA: lane<16 holds A[lane, 4k+{0,1}],
// lane>=16 holds A[lane-16, 4k+{2,3}].  B 4x16 assumed mirror-symmetric:
// lane<16 holds {B[4k+0,e],B[4k+1,e]}, lane>=16 holds {B[4k+2,e],B[4k+3,e]}
// (layout assumption is mechanical to flip; not observable compile-only).
// ---------------------------------------------------------------------------

#define E_TOTAL 160000
#define NN      10000
#define DIMA    128
#define DIMV    64
#define DIMD    32
#define RANK    32
#define R0F     8.0f

typedef __attribute__((ext_vector_type(2))) float v2f;
typedef __attribute__((ext_vector_type(8))) float v8f;

__device__ __forceinline__ v8f wmma4(v2f a, v2f b, v8f c) {
  // D(16x16 f32) = A(16x4 f32) x B(4x16 f32) + C
  return __builtin_amdgcn_wmma_f32_16x16x4_f32(false, a, false, b,
                                               (short)0, c, false, false);
}

template<int P> struct Pow3 { static constexpr int v = 3 * Pow3<P-1>::v; };
template<>      struct Pow3<0> { static constexpr int v = 1; };

struct TPArgs {
  const float* X;   // node features for this TP's left input
  const float* wl;  // (32, DIML) row-major
  const float* wr;  // (32, 8)    row-major
  const float* wo;  // (DOUT, 32) row-major
  float*       OUT; // output table base (B_a / B_v / B_d)
  int NS;           // node stride of OUT in floats
  int dnode;        // gather node (dst[e]) for this lane's edge column
  int snode;        // scatter node (src[e]) for this lane's edge column
};

template<int DIML, int DOUT, int PL, int PR, int PO>
__device__ __forceinline__ void process_tp(const TPArgs& A,
                                           const float* rad, const float* rh,
                                           float* m_lds, int lane) {
  constexpr int K2  = (PL + PR - PO) / 2;   // # contracted index pairs
  constexpr int CK  = Pow3<K2>::v;          // 3^k
  constexpr int NA  = Pow3<PL - K2>::v;     // surviving left spatial comps
  constexpr int PB  = Pow3<PR - K2>::v;     // surviving right spatial comps
  constexpr int CL3 = Pow3<PL>::v;          // spatial stride of left channels
  constexpr int PO3 = Pow3<PO>::v;          // output spatial size
  const int half = lane >> 4;               // which K-pair this lane supplies
  const int row  = lane & 15;               // A-frag row / D-tile column (edge)

  // --- g = wr @ rad^T : (rank x 16 edges), 2 M-tiles, K=8 in 2 steps ------
  v8f g0 = {}, g1 = {};
#pragma unroll
  for (int kk = 0; kk < 2; ++kk) {
    const int f0 = kk * 4 + half * 2;
    v2f bf; bf.x = rad[f0]; bf.y = rad[f0 + 1];
    v2f a0 = *(const v2f*)(A.wr + row * 8 + f0);
    v2f a1 = *(const v2f*)(A.wr + (16 + row) * 8 + f0);
    g0 = wmma4(a0, bf, g0);
    g1 = wmma4(a1, bf, g1);
  }

  const float* xn = A.X + (size_t)A.dnode * (DIML * CL3);

#pragma unroll
  for (int ap = 0; ap < NA; ++ap) {
    // --- m[:,ap] = sum_kidx (wl @ x[:, (ap,kidx)]) * rh^{ox k}[kidx] ------
    v8f m0 = {}, m1 = {};
#pragma unroll
    for (int kidx = 0; kidx < CK; ++kidx) {
      const int a = ap * CK + kidx;         // left spatial flat index
      v8f d0 = {}, d1 = {};
      for (int kk = 0; kk < DIML / 4; ++kk) {
        const int c0 = kk * 4 + half * 2;   // channel pair this lane supplies
        v2f bf;                             // B-frag: gathered node features
        bf.x = xn[c0 * CL3 + a];
        bf.y = xn[(c0 + 1) * CL3 + a];
        v2f a0 = *(const v2f*)(A.wl + row * DIML + c0);
        v2f a1 = *(const v2f*)(A.wl + (16 + row) * DIML + c0);
        d0 = wmma4(a0, bf, d0);
        d1 = wmma4(a1, bf, d1);
      }
      float rhk = 1.0f;                     // rh^{ox k}[kidx], lane-local (e = row)
      if constexpr (K2 == 1) rhk = rh[kidx];
      if constexpr (K2 == 2) rhk = rh[kidx / 3] * rh[kidx % 3];
#pragma unroll
      for (int v = 0; v < 8; ++v) { m0[v] += d0[v] * rhk; m1[v] += d1[v] * rhk; }
    }
    // scale by g (identical C/D register layout -> pure elementwise)
#pragma unroll
    for (int v = 0; v < 8; ++v) { m0[v] *= g0[v]; m1[v] *= g1[v]; }

    // --- LDS transpose: D layout (M=rank rows) -> B-frags with K=rank -----
#pragma unroll
    for (int v = 0; v < 8; ++v) {
      m_lds[(v + 8 * half) * 16 + row]      = m0[v];   // ranks 0..15 tile
      m_lds[(16 + v + 8 * half) * 16 + row] = m1[v];   // ranks 16..31 tile
    }
    // wave-private LDS; same-wave DS ops are in-order -> only stop the
    // compiler from reordering the cross-lane store->load pattern.
    asm volatile("" ::: "memory");

    // --- stage B: out[:,ap] = wo @ (g*m) : M=DOUT, K=rank=32 --------------
#pragma unroll
    for (int mt = 0; mt < DOUT / 16; ++mt) {
      v8f acc = {};
#pragma unroll
      for (int kk = 0; kk < 8; ++kk) {
        const int r0 = kk * 4 + half * 2;
        v2f bf; bf.x = m_lds[r0 * 16 + row]; bf.y = m_lds[(r0 + 1) * 16 + row];
        v2f af = *(const v2f*)(A.wo + (mt * 16 + row) * RANK + r0);
        acc = wmma4(af, bf, acc);
      }
      // outer product with rh^{ox (p_r-k)} and scatter-add at src node
      float* outn = A.OUT + (size_t)A.snode * A.NS;
#pragma unroll
      for (int bp = 0; bp < PB; ++bp) {
        float rhb = 1.0f;
        if constexpr (PR - K2 == 1) rhb = rh[bp];
        if constexpr (PR - K2 == 2) rhb = rh[bp / 3] * rh[bp % 3];
#pragma unroll
        for (int v = 0; v < 8; ++v) {
          const int o = mt * 16 + v + 8 * half;   // output channel (D row)
          atomicAdd(outn + o * PO3 + ap * PB + bp, acc[v] * rhb);
        }
      }
    }
    asm volatile("" ::: "memory");
  }
}

struct KParams {
  const float* r_ij;
  const float* x_a; const float* x_v; const float* x_d;
  const int* src; const int* dst;
  const float* wl[9]; const float* wr[9]; const float* wo[9];
  float* out_a; float* out_v; float* out_d;
};

__global__ __launch_bounds__(256) void msg_kernel(KParams P) {
  __shared__ float smem[8 * 512];           // 2 KB per wave (32x16 f32 tile)
  const int lane = threadIdx.x & 31;
  const int wave = threadIdx.x >> 5;
  float* m_lds = smem + wave * 512;

  const int tile = blockIdx.x * 8 + wave;   // one 16-edge tile per wave
  const int e    = tile * 16 + (lane & 15); // this lane's edge column

  const float rx = P.r_ij[3 * e + 0];
  const float ry = P.r_ij[3 * e + 1];
  const float rz = P.r_ij[3 * e + 2];
  const int snode = P.src[e];
  const int dnode = P.dst[e];

  // radial encoding: cos(pi*f*sqrt(|r|^2/r0)) * relu(1 - |r|^2/r0)
  const float xsq = (rx * rx + ry * ry + rz * rz) * (1.0f / R0F);
  const float env = fmaxf(1.0f - xsq, 0.0f);
  const float sr  = sqrtf(xsq);
  float rad[8];
#pragma unroll
  for (int f = 0; f < 8; ++f)
    rad[f] = cosf(3.14159265358979323846f * (float)f * sr) * env;

  // rh = tens_sigmoid(1, r * 17/r0)
  const float s17 = 17.0f / R0F;
  const float yx = rx * s17, yy = ry * s17, yz = rz * s17;
  const float nrm = sqrtf(yx * yx + yy * yy + yz * yz);
  const float sf  = (nrm > 1e-12f)
                      ? (2.0f / (1.0f + expf(-nrm)) - 1.0f) / nrm : 0.5f;
  float rh[3] = { yx * sf, yy * sf, yz * sf };

  TPArgs a; a.dnode = dnode; a.snode = snode;

  // TP_SPECS order: 000,101,202,011,110,112,211,220,222
  a.X=P.x_a; a.wl=P.wl[0]; a.wr=P.wr[0]; a.wo=P.wo[0]; a.OUT=P.out_a; a.NS=DIMA;
  process_tp<128,128,0,0,0>(a, rad, rh, m_lds, lane);
  a.X=P.x_v; a.wl=P.wl[1]; a.wr=P.wr[1]; a.wo=P.wo[1]; a.OUT=P.out_v; a.NS=DIMV*3;
  process_tp<64,64,1,0,1>(a, rad, rh, m_lds, lane);
  a.X=P.x_d; a.wl=P.wl[2]; a.wr=P.wr[2]; a.wo=P.wo[2]; a.OUT=P.out_d; a.NS=DIMD*9;
  process_tp<32,32,2,0,2>(a, rad, rh, m_lds, lane);
  a.X=P.x_a; a.wl=P.wl[3]; a.wr=P.wr[3]; a.wo=P.wo[3]; a.OUT=P.out_v; a.NS=DIMV*3;
  process_tp<128,64,0,1,1>(a, rad, rh, m_lds, lane);
  a.X=P.x_v; a.wl=P.wl[4]; a.wr=P.wr[4]; a.wo=P.wo[4]; a.OUT=P.out_a; a.NS=DIMA;
  process_tp<64,128,1,1,0>(a, rad, rh, m_lds, lane);
  a.X=P.x_v; a.wl=P.wl[5]; a.wr=P.wr[5]; a.wo=P.wo[5]; a.OUT=P.out_d; a.NS=DIMD*9;
  process_tp<64,32,1,1,2>(a, rad, rh, m_lds, lane);
  a.X=P.x_d; a.wl=P.wl[6]; a.wr=P.wr[6]; a.wo=P.wo[6]; a.OUT=P.out_v; a.NS=DIMV*3;
  process_tp<32,64,2,1,1>(a, rad, rh, m_lds, lane);
  a.X=P.x_d; a.wl=P.wl[7]; a.wr=P.wr[7]; a.wo=P.wo[7]; a.OUT=P.out_a; a.NS=DIMA;
  process_tp<32,128,2,2,0>(a, rad, rh, m_lds, lane);
  a.X=P.x_d; a.wl=P.wl[8]; a.wr=P.wr[8]; a.wo=P.wo[8]; a.OUT=P.out_d; a.NS=DIMD*9;
  process_tp<32,32,2,2,2>(a, rad, rh, m_lds, lane);
}

__global__ void zero_kernel(float* p, int n) {
  int i = blockIdx.x * blockDim.x + threadIdx.x;
  if (i < n) p[i] = 0.0f;
}

extern "C" void kernel_launch(void* const* d_in, const int* in_sizes, int n_in,
                              void* d_out, int out_size, void* d_ws, size_t ws_size,
                              hipStream_t stream) {
  KParams P;
  P.r_ij = (const float*)d_in[0];
  P.x_a  = (const float*)d_in[1];
  P.x_v  = (const float*)d_in[2];
  P.x_d  = (const float*)d_in[3];
  P.src  = (const int*)d_in[4];
  P.dst  = (const int*)d_in[5];
  // d_in[6] = n_nodes (compile-time constant NN)
  for (int i = 0; i < 9; ++i) {
    P.wl[i] = (const float*)d_in[7 + 3 * i];
    P.wr[i] = (const float*)d_in[8 + 3 * i];
    P.wo[i] = (const float*)d_in[9 + 3 * i];
  }
  float* out = (float*)d_out;
  P.out_a = out;                                // (NN,128)
  P.out_v = out + (size_t)NN * DIMA;            // (NN,64,3)
  P.out_d = out + (size_t)NN * (DIMA + DIMV*3); // (NN,32,3,3)

  const int total = NN * (DIMA + DIMV * 3 + DIMD * 9);
  zero_kernel<<<(total + 255) / 256, 256, 0, stream>>>(out, total);

  // 8 waves/block * 16 edges/wave = 128 edges/block; 160000/128 = 1250 exact
  msg_kernel<<<E_TOTAL / 128, 256, 0, stream>>>(P);
}